// CapLayer_22728966930492
// MI455X (gfx1250) — compile-verified
//
#include <hip/hip_runtime.h>

typedef __attribute__((ext_vector_type(2))) float v2f;
typedef __attribute__((ext_vector_type(8))) float v8f;

#define NS 32
#define J  10
#define DD 16
#define CIN 8
#define HW 36
#define NI 1152            // NS*HW input capsules
#define THREADS 512        // 16 waves of 32

__global__ __launch_bounds__(THREADS, 1)
void caps_fused(const float* __restrict__ x,
                const float* __restrict__ W,
                const float* __restrict__ bias,
                const float* __restrict__ b0,
                float* __restrict__ out)
{
    // LDS layout (total ~230 KB, fits CDNA5's 320 KB/workgroup)
    __shared__ float x_lds[NS*HW*16];   // [g*36+s][16]: c0..7 = x(g,c,s), c8 = 1.0, c9..15 = 0
    __shared__ float c_lds[NI*16];      // [i][16]: j0..9 = coupling coeffs, j10..15 = 0
    __shared__ float b_lds[J*NI];       // routing logits [j][i]
    __shared__ float y_lds[J*NS*16];    // [j][g][16]: c0..7 = y, c8 = csum
    __shared__ float u_lds[J*NS*16];    // [j][g][16]: c0..7 = u, c8 = bb
    __shared__ float s_lds[J*DD];
    __shared__ float v_lds[J*DD];

    const int tid = threadIdx.x;
    const int b   = blockIdx.x;
    const float* xb = x + (size_t)b * (NS*CIN*HW);

    // ---- stage x into LDS, transposed to [g][s][c] and augmented ----
    for (int t = tid; t < NS*HW; t += THREADS) {
        x_lds[t*16 + 8] = 1.0f;                 // ones column -> csum / bb terms
        #pragma unroll
        for (int c = 9; c < 16; ++c) x_lds[t*16 + c] = 0.0f;
    }
    for (int t = tid; t < NS*CIN*HW; t += THREADS) {  // coalesced read of x[b]
        int g = t / (CIN*HW);
        int r = t - g*(CIN*HW);
        int c = r / HW;
        int s = r - c*HW;
        x_lds[(g*HW + s)*16 + c] = xb[t];
    }
    for (int t = tid; t < NI; t += THREADS) {
        #pragma unroll
        for (int j = J; j < 16; ++j) c_lds[t*16 + j] = 0.0f;   // zero pad rows 10..15
    }
    for (int t = tid; t < J*NI; t += THREADS) b_lds[t] = b0[t];
    __syncthreads();

    const int lane  = tid & 31;
    const int wave  = tid >> 5;        // 0..15
    const int lhalf = lane & 15;       // m for A, n for B/D
    const int khi   = lane >> 4;       // half-wave select

    for (int it = 0; it < 3; ++it) {
        // ---- softmax over j for each input capsule i ----
        for (int i = tid; i < NI; i += THREADS) {
            float bv[J];
            float mx = -1e30f;
            #pragma unroll
            for (int j = 0; j < J; ++j) { bv[j] = b_lds[j*NI + i]; mx = fmaxf(mx, bv[j]); }
            float sum = 0.f;
            #pragma unroll
            for (int j = 0; j < J; ++j) { bv[j] = __expf(bv[j] - mx); sum += bv[j]; }
            float inv = 1.0f / sum;
            #pragma unroll
            for (int j = 0; j < J; ++j) c_lds[i*16 + j] = bv[j] * inv;
        }
        __syncthreads();

        // ---- y[j,g,c] = sum_s c[j,(g,s)] * x[g,c,s] via WMMA f32 16x16x4 ----
        // Per group g: Y(16x16) = Cmat(16x36) @ Xaug(36x16), K = 36 = 9 k-steps.
        #pragma unroll
        for (int gg = 0; gg < 2; ++gg) {
            int g = wave*2 + gg;
            v8f acc = {};
            for (int kk = 0; kk < HW; kk += 4) {
                // A[m][k] = c_lds[(g*36+k)*16 + m]; B[k][n] = x_lds[(g*36+k)*16 + n]
                int k0 = (g*HW + kk + 2*khi)*16 + lhalf;
                v2f a, bm;
                a.x  = c_lds[k0];
                a.y  = c_lds[k0 + 16];
                bm.x = x_lds[k0];
                bm.y = x_lds[k0 + 16];
                acc = __builtin_amdgcn_wmma_f32_16x16x4_f32(
                        false, a, false, bm, (short)0, acc, false, false);
            }
            #pragma unroll
            for (int r = 0; r < 8; ++r) {
                int m = r + 8*khi;                       // output row j
                if (m < J) y_lds[(m*NS + g)*16 + lhalf] = acc[r];
            }
        }
        __syncthreads();

        // ---- s[j,d] = sum_{g,c} y*W + csum*bias (W/bias stream from L2) ----
        if (tid < J*DD) {
            int j = tid >> 4, d = tid & 15;
            float acc = 0.f;
            for (int g = 0; g < NS; ++g) {
                const float* wp = W + ((size_t)g*(J*DD) + j*DD + d)*CIN;
                const float* yp = y_lds + (j*NS + g)*16;
                #pragma unroll
                for (int c = 0; c < CIN; ++c) acc += yp[c] * wp[c];
                acc += yp[8] * bias[g*(J*DD) + j*DD + d];
            }
            s_lds[tid] = acc;
        }
        __syncthreads();

        // ---- squash: v = s * ||s|| / (1 + ||s||^2) ----
        if (tid < J*DD) {
            int j = tid >> 4;
            float n2 = 0.f;
            #pragma unroll
            for (int d = 0; d < DD; ++d) { float t = s_lds[j*DD + d]; n2 += t*t; }
            float n  = sqrtf(n2);
            float vv = s_lds[tid] * (n / (1.0f + n2));
            v_lds[tid] = vv;
            if (it == 2) out[(size_t)b*(J*DD) + tid] = vv;
        }
        __syncthreads();

        if (it == 2) break;   // last iteration's b-update is dead in the reference

        // ---- u[j,g,c] = sum_d v[j,d]*W[g,jd,c];  u[..,8] = sum_d v*bias ----
        if (tid < J*NS) {
            int j = tid >> 5, g = tid & 31;
            const float* wb = W + (size_t)g*(J*DD)*CIN + j*DD*CIN;
            float u[9];
            #pragma unroll
            for (int c = 0; c < 9; ++c) u[c] = 0.f;
            #pragma unroll
            for (int d = 0; d < DD; ++d) {
                float vv = v_lds[j*DD + d];
                const float* wr = wb + d*CIN;
                #pragma unroll
                for (int c = 0; c < CIN; ++c) u[c] += vv * wr[c];
                u[8] += vv * bias[g*(J*DD) + j*DD + d];
            }
            #pragma unroll
            for (int c = 0; c < 9; ++c) u_lds[(j*NS + g)*16 + c] = u[c];
        }
        __syncthreads();

        // ---- b[j,i] += sum_c xaug[g,c,s] * u[j,g,c]  (ones col supplies bb) ----
        for (int i = tid; i < NI; i += THREADS) {
            int g = i / HW;
            const float* xp = x_lds + i*16;
            #pragma unroll
            for (int j = 0; j < J; ++j) {
                const float* up = u_lds + (j*NS + g)*16;
                float db = 0.f;
                #pragma unroll
                for (int c = 0; c < 9; ++c) db += xp[c] * up[c];
                b_lds[j*NI + i] += db;
            }
        }
        __syncthreads();
    }
}

extern "C" void kernel_launch(void* const* d_in, const int* in_sizes, int n_in,
                              void* d_out, int out_size, void* d_ws, size_t ws_size,
                              hipStream_t stream) {
    const float* x    = (const float*)d_in[0];
    const float* W    = (const float*)d_in[1];
    const float* bias = (const float*)d_in[2];
    const float* b0   = (const float*)d_in[3];
    float* out = (float*)d_out;
    int bs = in_sizes[0] / (NS*CIN*HW);   // 256
    caps_fused<<<bs, THREADS, 0, stream>>>(x, W, bias, b0, out);
}